// forwardLayerWF_88562225643987
// MI455X (gfx1250) — compile-verified
//
#include <hip/hip_runtime.h>

typedef __attribute__((ext_vector_type(2))) float v2f;
typedef __attribute__((ext_vector_type(8))) float v8f;

#define TWO_PI_F 6.28318530717958647692f
#define KC 32          // K-chunk staged in LDS
#define LDA_S 33       // padded LDS row stride for A (coprime with 64 banks)
#define LDB_S 66       // padded LDS row stride for B

// ---------------------------------------------------------------------------
// gfx1250 async global->LDS copy (ASYNCcnt path), with portable fallback.
// Builtin signature (probe-confirmed present): takes address-space-qualified
// int* pointers: (global int* src, local int* dst, imm offset, imm cpol).
// ---------------------------------------------------------------------------
__device__ inline void async_copy_f32(float* lds_dst, const float* gsrc) {
#if __has_builtin(__builtin_amdgcn_global_load_async_to_lds_b32)
  __builtin_amdgcn_global_load_async_to_lds_b32(
      (__attribute__((address_space(1))) int*)(float*)gsrc,
      (__attribute__((address_space(3))) int*)lds_dst, 0, 0);
#else
  *lds_dst = *gsrc;
#endif
}

__device__ inline void async_wait_all() {
#if __has_builtin(__builtin_amdgcn_global_load_async_to_lds_b32)
#if __has_builtin(__builtin_amdgcn_s_wait_asynccnt)
  __builtin_amdgcn_s_wait_asynccnt(0);
#else
  asm volatile("s_wait_asynccnt 0" ::: "memory");
#endif
#endif
}

__device__ inline v8f wmma4(v2f a, v2f b, v8f c) {
  return __builtin_amdgcn_wmma_f32_16x16x4_f32(false, a, false, b,
                                               (short)0, c, false, false);
}

// ---------------------------------------------------------------------------
// Twiddle-matrix fill: W[r][c] with fftshift baked in.
//   shift_on_row=1:  phase index f = (row - N/2) * col   (forward-shifted DFT)
//   shift_on_row=0:  phase index f = (col - N/2) * row   (inverse-shifted DFT)
//   transpose=1:     stores the transposed matrix.
// Exact integer mod-N reduction keeps float sincos accurate.
// ---------------------------------------------------------------------------
__global__ void fill_twiddle_k(float* __restrict__ wr, float* __restrict__ wi,
                               int N, float sgn, float scale,
                               int shift_on_row, int transpose) {
  int idx = blockIdx.x * blockDim.x + threadIdx.x;
  if (idx >= N * N) return;
  int r = idx / N, c = idx - r * N;
  long long a = transpose ? c : r;
  long long b = transpose ? r : c;
  long long f = shift_on_row ? (a - N / 2) * b : (b - N / 2) * a;
  int fm = (int)(f % N);
  if (fm < 0) fm += N;
  float ang = sgn * TWO_PI_F * (float)fm / (float)N;
  float s, cc;
  __sincosf(ang, &s, &cc);
  wr[idx] = scale * cc;
  wi[idx] = scale * s;
}

// ---------------------------------------------------------------------------
// Complex GEMM C = A*B via V_WMMA_F32_16X16X4_F32, LDS-tiled.
// 128 threads = 4 waves; workgroup computes a 64x64 C macro-tile.
// Wave (wr,wc) owns a 32x32 quadrant = 2x2 WMMA subtiles.
// A(64xKC) and B(KCx64) complex panels are staged in LDS per K-chunk using
// async global->LDS loads; inner loop reads fragments from LDS only.
// Cr = accRR - accII,  Ci = accRI + accIR (chained accumulator).
// a_col_mode: offset A columns by kxl = 192 + Ns[2*step]   (window rows)
// b_row_mode: offset B rows    by kyl = 192 + Ns[2*step+1] (window cols)
// out_interleaved: write (re,im) pairs (final gradient output), else SoA.
// No divergence anywhere -> EXEC all-1s for every WMMA.
// ---------------------------------------------------------------------------
__global__ __launch_bounds__(128) void cgemm_wmma_k(
    const float* __restrict__ Ar, const float* __restrict__ Ai, int lda, long sA,
    const float* __restrict__ Br, const float* __restrict__ Bi, int ldb, long sB,
    float* __restrict__ Cr_p, float* __restrict__ Ci_p, long sC, int ldc,
    int M, int N, int K, float out_scale, int out_interleaved,
    const int* __restrict__ Ns, int step, int a_col_mode, int b_row_mode) {
  __shared__ float lAr[64 * LDA_S], lAi[64 * LDA_S];
  __shared__ float lBr[KC * LDB_S], lBi[KC * LDB_S];

  int tilesN = N >> 6;
  int bt = blockIdx.x;
  int btm = bt / tilesN, btn = bt - btm * tilesN;
  int batch = blockIdx.y;
  int tid = threadIdx.x;
  int wave = tid >> 5, lane = tid & 31;
  int wr = wave >> 1, wc = wave & 1;
  int hv = lane >> 4, lm = lane & 15;

  int aoff = a_col_mode ? (192 + Ns[2 * step]) : 0;
  int boff = b_row_mode ? (192 + Ns[2 * step + 1]) * ldb : 0;

  const float* Aer = Ar + (long)batch * sA + aoff;
  const float* Aei = Ai + (long)batch * sA + aoff;
  const float* Ber = Br + (long)batch * sB + boff;
  const float* Bei = Bi + (long)batch * sB + boff;

  int rowBase = btm * 64, colBase = btn * 64;

  v8f accRR[2][2] = {{{}, {}}, {{}, {}}};
  v8f accII[2][2] = {{{}, {}}, {{}, {}}};
  v8f accCI[2][2] = {{{}, {}}, {{}, {}}};

  for (int k0 = 0; k0 < K; k0 += KC) {
    // ---- stage A (64 x KC) and B (KC x 64) complex panels into LDS ----
    for (int e = tid; e < 64 * KC; e += 128) {
      int row = e >> 5;          // KC == 32
      int kk = e & (KC - 1);
      long ga = (long)(rowBase + row) * lda + k0 + kk;
      int lo = row * LDA_S + kk;
      async_copy_f32(&lAr[lo], &Aer[ga]);
      async_copy_f32(&lAi[lo], &Aei[ga]);
    }
    for (int e = tid; e < KC * 64; e += 128) {
      int kk = e >> 6;
      int col = e & 63;
      long gb = (long)(k0 + kk) * ldb + colBase + col;
      int lo = kk * LDB_S + col;
      async_copy_f32(&lBr[lo], &Ber[gb]);
      async_copy_f32(&lBi[lo], &Bei[gb]);
    }
    if (k0 + KC < K) {  // hint next chunk toward L2 while we compute
      __builtin_prefetch(&Aer[(long)(rowBase + (tid >> 1)) * lda + k0 + KC], 0, 1);
      __builtin_prefetch(&Ber[(long)(k0 + KC + (tid & 31)) * ldb + colBase], 0, 1);
    }
    async_wait_all();
    __syncthreads();

    // ---- multiply from LDS ----
    for (int kk = 0; kk < KC; kk += 4) {
      v2f arf[2], aif[2], brf[2], bif[2];
#pragma unroll
      for (int sm = 0; sm < 2; ++sm) {
        int o = (wr * 32 + sm * 16 + lm) * LDA_S + kk + hv * 2;
        arf[sm].x = lAr[o]; arf[sm].y = lAr[o + 1];
        aif[sm].x = lAi[o]; aif[sm].y = lAi[o + 1];
      }
#pragma unroll
      for (int sn = 0; sn < 2; ++sn) {
        int o0 = (kk + hv * 2) * LDB_S + wc * 32 + sn * 16 + lm;
        brf[sn].x = lBr[o0]; brf[sn].y = lBr[o0 + LDB_S];
        bif[sn].x = lBi[o0]; bif[sn].y = lBi[o0 + LDB_S];
      }
#pragma unroll
      for (int sm = 0; sm < 2; ++sm)
#pragma unroll
        for (int sn = 0; sn < 2; ++sn) {
          accRR[sm][sn] = wmma4(arf[sm], brf[sn], accRR[sm][sn]);
          accII[sm][sn] = wmma4(aif[sm], bif[sn], accII[sm][sn]);
          accCI[sm][sn] = wmma4(arf[sm], bif[sn], accCI[sm][sn]);
          accCI[sm][sn] = wmma4(aif[sm], brf[sn], accCI[sm][sn]);
        }
    }
    __syncthreads();
  }

  // ---- write out ----
#pragma unroll
  for (int sm = 0; sm < 2; ++sm)
#pragma unroll
    for (int sn = 0; sn < 2; ++sn) {
      int n = colBase + wc * 32 + sn * 16 + lm;
      if (out_interleaved) {
        float* out = Cr_p + (long)batch * sC;
#pragma unroll
        for (int j = 0; j < 8; ++j) {
          int m = rowBase + wr * 32 + sm * 16 + j + 8 * hv;  // VGPR j -> rows j, j+8
          long o = ((long)m * ldc + n) * 2;
          out[o]     = (accRR[sm][sn][j] - accII[sm][sn][j]) * out_scale;
          out[o + 1] = accCI[sm][sn][j] * out_scale;
        }
      } else {
        float* crp = Cr_p + (long)batch * sC;
        float* cip = Ci_p + (long)batch * sC;
#pragma unroll
        for (int j = 0; j < 8; ++j) {
          int m = rowBase + wr * 32 + sm * 16 + j + 8 * hv;
          long o = (long)m * ldc + n;
          crp[o] = (accRR[sm][sn][j] - accII[sm][sn][j]) * out_scale;
          cip[o] = accCI[sm][sn][j] * out_scale;
        }
      }
    }
}

// ---------------------------------------------------------------------------
// Pointwise kernels (all operate on [b=2, 128, 128] = 32768 elems).
// ---------------------------------------------------------------------------
__global__ void extract_psi_k(const float* __restrict__ Sr, const float* __restrict__ Si,
                              const float* __restrict__ P,
                              float* __restrict__ pr, float* __restrict__ pi,
                              const int* __restrict__ Ns, int step) {
  int idx = blockIdx.x * blockDim.x + threadIdx.x;
  if (idx >= 2 * 128 * 128) return;
  int b = idx >> 14;
  int pix = idx & 16383;
  int u = pix >> 7, v = pix & 127;
  int kxl = 192 + Ns[2 * step], kyl = 192 + Ns[2 * step + 1];
  long so = (long)b * 262144 + (long)(kxl + u) * 512 + (kyl + v);
  float p = P[pix];
  pr[idx] = Sr[so] * p;
  pi[idx] = Si[so] * p;
}

__global__ void nonlinear_k(float* __restrict__ vr, float* __restrict__ vi,
                            const float* __restrict__ I, int step) {
  int idx = blockIdx.x * blockDim.x + threadIdx.x;
  if (idx >= 2 * 128 * 128) return;
  int b = idx >> 14;
  int pix = idx & 16383;
  float re = vr[idx], im = vi[idx];
  float Ik = I[(long)b * 49 * 16384 + (long)step * 16384 + pix];
  float f = re * re + im * im - Ik;
  vr[idx] = f * re;
  vi[idx] = f * im;
}

// patch *= P; then spectrum update S -= MU*scale_bwd*patch (window only).
__global__ void applyP_updateS_k(float* __restrict__ pr, float* __restrict__ pi,
                                 const float* __restrict__ P,
                                 float* __restrict__ Sr, float* __restrict__ Si,
                                 const int* __restrict__ Ns, int step) {
  int idx = blockIdx.x * blockDim.x + threadIdx.x;
  if (idx >= 2 * 128 * 128) return;
  int b = idx >> 14;
  int pix = idx & 16383;
  int u = pix >> 7, v = pix & 127;
  int kxl = 192 + Ns[2 * step], kyl = 192 + Ns[2 * step + 1];
  float p = P[pix];
  float xr = pr[idx] * p, xi = pi[idx] * p;
  pr[idx] = xr;
  pi[idx] = xi;
  long so = (long)b * 262144 + (long)(kxl + u) * 512 + (kyl + v);
  const float c = 0.16f;  // MU * scale_bwd = 0.01 * 16
  Sr[so] -= c * xr;
  Si[so] -= c * xi;
}

// ---------------------------------------------------------------------------
// Host driver
// ---------------------------------------------------------------------------
extern "C" void kernel_launch(void* const* d_in, const int* in_sizes, int n_in,
                              void* d_out, int out_size, void* d_ws, size_t ws_size,
                              hipStream_t stream) {
  (void)in_sizes; (void)n_in; (void)out_size;
  const float* x_real = (const float*)d_in[0];   // [2,512,512]
  const float* x_imag = (const float*)d_in[1];   // [2,512,512]
  const float* P      = (const float*)d_in[2];   // [128,128]
  const float* I      = (const float*)d_in[3];   // [2,49,128,128]
  const int*   Ns     = (const int*)d_in[4];     // [49,2]
  float* out = (float*)d_out;                    // [2,49,512,512] complex interleaved

  const size_t N512 = 512 * 512, N128 = 128 * 128;
  float* w = (float*)d_ws;
  size_t o = 0;
  auto alloc = [&](size_t n) { float* p = w + o; o += n; return p; };
  float *Wf512r = alloc(N512), *Wf512i = alloc(N512);
  float *WfT512r = alloc(N512), *WfT512i = alloc(N512);
  float *Wi512r = alloc(N512), *Wi512i = alloc(N512);
  float *WiT512r = alloc(N512), *WiT512i = alloc(N512);
  float *Sr = alloc(2 * N512), *Si = alloc(2 * N512);
  float *Tr = alloc(2 * N512), *Ti = alloc(2 * N512);
  float *Wi128r = alloc(N128), *Wi128i = alloc(N128);
  float *WiT128r = alloc(N128), *WiT128i = alloc(N128);
  float *Wf128r = alloc(N128), *Wf128i = alloc(N128);
  float *WfT128r = alloc(N128), *WfT128i = alloc(N128);
  float *bAr = alloc(2 * N128), *bAi = alloc(2 * N128);
  float *bBr = alloc(2 * N128), *bBi = alloc(2 * N128);
  if (ws_size < o * sizeof(float)) return;

  // --- twiddle matrices ---
  {
    dim3 b(256), g512((N512 + 255) / 256), g128((N128 + 255) / 256);
    hipLaunchKernelGGL(fill_twiddle_k, g512, b, 0, stream, Wf512r, Wf512i, 512, -1.0f, 1.0f, 1, 0);
    hipLaunchKernelGGL(fill_twiddle_k, g512, b, 0, stream, WfT512r, WfT512i, 512, -1.0f, 1.0f, 1, 1);
    hipLaunchKernelGGL(fill_twiddle_k, g512, b, 0, stream, Wi512r, Wi512i, 512, 1.0f, 1.0f / 512.0f, 0, 0);
    hipLaunchKernelGGL(fill_twiddle_k, g512, b, 0, stream, WiT512r, WiT512i, 512, 1.0f, 1.0f / 512.0f, 0, 1);
    // 128-pt inverse with scale_fwd = 1/16 folded in: (1/128)*(1/4) per axis
    hipLaunchKernelGGL(fill_twiddle_k, g128, b, 0, stream, Wi128r, Wi128i, 128, 1.0f, 1.0f / 512.0f, 0, 0);
    hipLaunchKernelGGL(fill_twiddle_k, g128, b, 0, stream, WiT128r, WiT128i, 128, 1.0f, 1.0f / 512.0f, 0, 1);
    hipLaunchKernelGGL(fill_twiddle_k, g128, b, 0, stream, Wf128r, Wf128i, 128, -1.0f, 1.0f, 1, 0);
    hipLaunchKernelGGL(fill_twiddle_k, g128, b, 0, stream, WfT128r, WfT128i, 128, -1.0f, 1.0f, 1, 1);
  }

  auto cgemm = [&](const float* Ar, const float* Ai, int lda, long sA, int aMode,
                   const float* Br, const float* Bi, int ldb, long sB, int bMode,
                   float* Crp, float* Cip, long sC, int ldc, int interleaved,
                   int M, int N, int K, float scale, int step) {
    int tiles = (M / 64) * (N / 64);
    dim3 g(tiles, 2), b(128);
    hipLaunchKernelGGL(cgemm_wmma_k, g, b, 0, stream,
                       Ar, Ai, lda, sA, Br, Bi, ldb, sB, Crp, Cip, sC, ldc,
                       M, N, K, scale, interleaved, Ns, step, aMode, bMode);
  };

  // --- init: S = fftshift(fft2(x)) = Wf512 * x * Wf512^T (batched over b) ---
  cgemm(Wf512r, Wf512i, 512, 0, 0, x_real, x_imag, 512, (long)N512, 0,
        Tr, Ti, (long)N512, 512, 0, 512, 512, 512, 1.0f, 0);
  cgemm(Tr, Ti, 512, (long)N512, 0, WfT512r, WfT512i, 512, 0, 0,
        Sr, Si, (long)N512, 512, 0, 512, 512, 512, 1.0f, 0);

  // --- scan over 49 illumination positions ---
  dim3 pb(256), pg((2 * 128 * 128 + 255) / 256);
  for (int k = 0; k < 49; ++k) {
    // psi = S[window] * P
    hipLaunchKernelGGL(extract_psi_k, pg, pb, 0, stream, Sr, Si, P, bAr, bAi, Ns, k);
    // inv = Wi128 * psi * Wi128^T   (ifft2(ifftshift(.)) * scale_fwd)
    cgemm(Wi128r, Wi128i, 128, 0, 0, bAr, bAi, 128, (long)N128, 0,
          bBr, bBi, (long)N128, 128, 0, 128, 128, 128, 1.0f, k);
    cgemm(bBr, bBi, 128, (long)N128, 0, WiT128r, WiT128i, 128, 0, 0,
          bAr, bAi, (long)N128, 128, 0, 128, 128, 128, 1.0f, k);
    // diff = (|inv|^2 - I_k) * inv
    hipLaunchKernelGGL(nonlinear_k, pg, pb, 0, stream, bAr, bAi, I, k);
    // patch = Wf128 * diff * Wf128^T   (fftshift(fft2(diff)))
    cgemm(Wf128r, Wf128i, 128, 0, 0, bAr, bAi, 128, (long)N128, 0,
          bBr, bBi, (long)N128, 128, 0, 128, 128, 128, 1.0f, k);
    cgemm(bBr, bBi, 128, (long)N128, 0, WfT128r, WfT128i, 128, 0, 0,
          bAr, bAi, (long)N128, 128, 0, 128, 128, 128, 1.0f, k);
    // patch *= P ;  S[window] -= MU*scale_bwd*patch  (spectral-domain update)
    hipLaunchKernelGGL(applyP_updateS_k, pg, pb, 0, stream, bAr, bAi, P, Sr, Si, Ns, k);
    // sparse 512 IDFT of zero-padded patch:
    //   T = patch * WiT512[kyl: , :]   (128 x 512, K=128)
    cgemm(bAr, bAi, 128, (long)N128, 0, WiT512r, WiT512i, 512, 0, /*kyl rows*/ 1,
          Tr, Ti, (long)N512, 512, 0, 128, 512, 128, 1.0f, k);
    //   invZ = Wi512[:, kxl:] * T  -> write gradients (interleaved), *scale_bwd
    cgemm(Wi512r, Wi512i, 512, 0, /*kxl cols*/ 1, Tr, Ti, 512, (long)N512, 0,
          out + (long)k * N512 * 2, nullptr, (long)49 * N512 * 2, 512, 1,
          512, 512, 128, 16.0f, k);
  }
}